// RNN_Exp_Decay_60808146977436
// MI455X (gfx1250) — compile-verified
//
#include <hip/hip_runtime.h>
#include <cmath>

// RNN exp-decay GRU (4 instances: dp_fw, dp_bw, cp_fw, cp_bw), H=16, T=2000, B=1024.
// MI455X / gfx1250 (wave32) strategy:
//  - One wave per (16-row batch tile, GRU instance). grid = (B/16, 4) = 256 waves.
//  - Per step: xW = E @ W_ih^T and hw = h_dec @ W_hh^T as 16x48x16 fp32 GEMMs via
//    chained V_WMMA_F32_16X16X4_F32 (4 K-chunks x 3 N-tiles each = 24 WMMA/step).
//  - Weights resident in VGPRs in WMMA B-layout (loaded once).
//  - Embedding gather software-pipelined: A-operands for step s+1 load during step s;
//    index for step s+2 loads one level deeper (covers the idx->embed dependent chain).
//  - Native transcendentals (v_exp_f32 / v_rcp_f32) for decay/sigmoid/tanh to shorten
//    the serial per-step critical path.
//  - dt broadcast via wave shuffle (ds_bpermute); h relayout D->A via 1KB LDS.

typedef __attribute__((ext_vector_type(2))) float v2f;
typedef __attribute__((ext_vector_type(4))) float v4f;
typedef __attribute__((ext_vector_type(8))) float v8f;

#define NB 1024
#define NT 2000

struct GruArgs {
  const float* embed[2];   // dp, cp embedding tables [65536,16]
  const int*   idx[2];     // dp, cp indices [B,T]
  const float* times[2];   // dp, cp absolute times [B,T]
  const float* W_ih[4];    // [48,16]
  const float* W_hh[4];    // [48,16]
  const float* b_ih[4];    // [48]
  const float* b_hh[4];    // [48]
  const float* dec[4];     // [16]
  float*       hT;         // workspace: [4][B][16] final hidden states
};

__device__ __forceinline__ v8f wmma_f32_k4(v2f a, v2f b, v8f c) {
  // D(16x16,f32) += A(16x4,f32) x B(4x16,f32); 8-arg VOP3P form.
  return __builtin_amdgcn_wmma_f32_16x16x4_f32(false, a, false, b, (short)0, c,
                                               false, false);
}

// Native-HW transcendentals: v_exp_f32 + v_rcp_f32 (saturate correctly at +-inf).
__device__ __forceinline__ float fast_sigmoid(float x) {
  return __builtin_amdgcn_rcpf(1.f + __expf(-x));
}
__device__ __forceinline__ float fast_tanh(float x) {
  return 1.f - 2.f * __builtin_amdgcn_rcpf(1.f + __expf(2.f * x));
}
__device__ __forceinline__ float round3(float x) { return rintf(x * 1000.f) * 0.001f; }

__global__ __launch_bounds__(32)
void gru_scan_kernel(GruArgs ga) {
  const int gru  = blockIdx.y;        // 0:dp_fw 1:dp_bw 2:cp_fw 3:cp_bw
  const int feat = gru >> 1;          // 0=dp, 1=cp
  const bool bw  = (gru & 1) != 0;
  const int b0   = blockIdx.x * 16;   // batch tile start
  const int lane = threadIdx.x;       // 0..31
  const int m    = lane & 15;         // A-layout row / batch row within tile
  const int half = lane >> 4;         // 0|1: selects K pair in A/B layout
  const int n    = m;                 // C/D-layout column (hidden index)

  const float* __restrict__ emb = ga.embed[feat];
  const int*   __restrict__ idx = ga.idx[feat];
  const float* __restrict__ tms = ga.times[feat];
  const float* __restrict__ Wih = ga.W_ih[gru];
  const float* __restrict__ Whh = ga.W_hh[gru];

  __shared__ __align__(16) float ht_lds[16 * 16];  // h transposed: [n][m]

  // Per-lane scalars.
  float bih[3], bhh[3];
#pragma unroll
  for (int t3 = 0; t3 < 3; ++t3) {
    bih[t3] = ga.b_ih[gru][t3 * 16 + n];
    bhh[t3] = ga.b_hh[gru][t3 * 16 + n];
  }
  const float decn = ga.dec[gru][n];

  // Preload B operands (W^T) in WMMA B layout; resident in VGPRs for whole scan.
  // Chunk c covers K rows 4c..4c+3; lane holds (k = 4c+2*half+{0,1}, col g = t3*16+n).
  v2f Bih[3][4], Bhh[3][4];
#pragma unroll
  for (int t3 = 0; t3 < 3; ++t3) {
#pragma unroll
    for (int c = 0; c < 4; ++c) {
      const int g = t3 * 16 + n;
      const int k = 4 * c + 2 * half;
      v2f wi; wi.x = Wih[g * 16 + k]; wi.y = Wih[g * 16 + k + 1];
      v2f wh; wh.x = Whh[g * 16 + k]; wh.y = Whh[g * 16 + k + 1];
      Bih[t3][c] = wi;
      Bhh[t3][c] = wh;
    }
  }

  const long rbase = (long)(b0 + m) * NT;

  // -------- software pipeline prologue --------
  // Invariant at top of iteration s: AeN/tjN/tpN hold step-s operands,
  // eidxN holds the embedding index for step s+1.
  v2f   AeN[4];
  float tjN, tpN;
  int   eidxN;
  {
    const int j0 = bw ? (NT - 1) : 0;
    const int e0 = idx[rbase + j0];
    const float* __restrict__ er = emb + (long)e0 * 16;
#pragma unroll
    for (int c = 0; c < 4; ++c) {
      const int k = 4 * c + 2 * half;
      v2f a; a.x = er[k]; a.y = er[k + 1];
      AeN[c] = a;
    }
    tjN = tms[rbase + j0];
    tpN = (j0 > 0) ? tms[rbase + j0 - 1] : tjN;
    const int j1 = bw ? (NT - 2) : 1;
    eidxN = idx[rbase + j1];
  }

  // h in C/D layout: element r of v8f -> (M = r + 8*half, N = n).
  v8f h = {0.f, 0.f, 0.f, 0.f, 0.f, 0.f, 0.f, 0.f};

  for (int s = 0; s < NT; ++s) {
    // ---- consume pipelined operands for step s ----
    v2f Ae[4];
#pragma unroll
    for (int c = 0; c < 4; ++c) Ae[c] = AeN[c];
    const float dtv = round3(tjN) - round3(tpN);  // 0 at sequence start (tp==tj)

    // ---- issue loads for steps s+1 (embed, times) and s+2 (idx) ----
    {
      const int j1 = bw ? max(NT - 2 - s, 0) : min(s + 1, NT - 1);
      const int j2 = bw ? max(NT - 3 - s, 0) : min(s + 2, NT - 1);
      const float* __restrict__ er = emb + (long)eidxN * 16;
#pragma unroll
      for (int c = 0; c < 4; ++c) {
        const int k = 4 * c + 2 * half;
        v2f a; a.x = er[k]; a.y = er[k + 1];
        AeN[c] = a;
      }
      tjN = tms[rbase + j1];
      tpN = (j1 > 0) ? tms[rbase + j1 - 1] : tjN;
      eidxN = idx[rbase + j2];
    }

    // ---- input transform: x{r,z,n} = E @ W_ih^T ----
    v8f xr = {0.f,0.f,0.f,0.f,0.f,0.f,0.f,0.f};
    v8f xz = {0.f,0.f,0.f,0.f,0.f,0.f,0.f,0.f};
    v8f xn = {0.f,0.f,0.f,0.f,0.f,0.f,0.f,0.f};
#pragma unroll
    for (int c = 0; c < 4; ++c) {
      xr = wmma_f32_k4(Ae[c], Bih[0][c], xr);
      xz = wmma_f32_k4(Ae[c], Bih[1][c], xz);
      xn = wmma_f32_k4(Ae[c], Bih[2][c], xn);
    }

    // ---- exponential decay: hd = h * exp(-relu(dec[n]*dt[m])), dt via shuffle ----
    v8f hd;
#pragma unroll
    for (int r = 0; r < 8; ++r) {
      const float dtm = __shfl(dtv, r + 8 * half, 32);  // row M = r + 8*half
      hd[r] = h[r] * __expf(-fmaxf(decn * dtm, 0.f));
    }

    // ---- stage hd (C/D layout) -> LDS transposed [n][m] ----
    __syncthreads();  // prior iteration's A-layout reads complete
    {
      v4f lo; lo.x = hd[0]; lo.y = hd[1]; lo.z = hd[2]; lo.w = hd[3];
      v4f hi; hi.x = hd[4]; hi.y = hd[5]; hi.z = hd[6]; hi.w = hd[7];
      *(v4f*)&ht_lds[n * 16 + 8 * half]     = lo;
      *(v4f*)&ht_lds[n * 16 + 8 * half + 4] = hi;
    }
    __syncthreads();

    // ---- reload hd in WMMA A layout ----
    v2f Ah[4];
#pragma unroll
    for (int c = 0; c < 4; ++c) {
      const int k = 4 * c + 2 * half;
      v2f a; a.x = ht_lds[k * 16 + m]; a.y = ht_lds[(k + 1) * 16 + m];
      Ah[c] = a;
    }

    // ---- hidden transform: h{r,z,n} = hd @ W_hh^T ----
    v8f hr = {0.f,0.f,0.f,0.f,0.f,0.f,0.f,0.f};
    v8f hz = {0.f,0.f,0.f,0.f,0.f,0.f,0.f,0.f};
    v8f hn = {0.f,0.f,0.f,0.f,0.f,0.f,0.f,0.f};
#pragma unroll
    for (int c = 0; c < 4; ++c) {
      hr = wmma_f32_k4(Ah[c], Bhh[0][c], hr);
      hz = wmma_f32_k4(Ah[c], Bhh[1][c], hz);
      hn = wmma_f32_k4(Ah[c], Bhh[2][c], hn);
    }

    // ---- GRU gates (pointwise, C/D layout) ----
#pragma unroll
    for (int r = 0; r < 8; ++r) {
      const float rr = fast_sigmoid(xr[r] + bih[0] + hr[r] + bhh[0]);
      const float zz = fast_sigmoid(xz[r] + bih[1] + hz[r] + bhh[1]);
      const float nn = fast_tanh(xn[r] + bih[2] + rr * (hn[r] + bhh[2]));
      h[r] = (1.f - zz) * nn + zz * hd[r];
    }
  }

  // ---- write final hidden state: hT[gru][b0+M][n] ----
#pragma unroll
  for (int r = 0; r < 8; ++r) {
    const int M = r + 8 * half;
    ga.hT[((long)gru * NB + b0 + M) * 16 + n] = h[r];
  }
}

__global__ __launch_bounds__(256)
void head_kernel(const float* __restrict__ hT, const float* __restrict__ statics,
                 const float* __restrict__ fc_dp_w, const float* __restrict__ fc_dp_b,
                 const float* __restrict__ fc_cp_w, const float* __restrict__ fc_cp_b,
                 const float* __restrict__ fc_fin_w, const float* __restrict__ fc_fin_b,
                 float* __restrict__ out) {
  const int b = blockIdx.x * blockDim.x + threadIdx.x;
  if (b >= NB) return;
  const float* hdf = hT + (0L * NB + b) * 16;
  const float* hdb = hT + (1L * NB + b) * 16;
  const float* hcf = hT + (2L * NB + b) * 16;
  const float* hcb = hT + (3L * NB + b) * 16;
  float dpv = fc_dp_b[0], cpv = fc_cp_b[0];
#pragma unroll
  for (int k = 0; k < 16; ++k) {
    dpv = fmaf(fc_dp_w[k], hdf[k], fmaf(fc_dp_w[16 + k], hdb[k], dpv));
    cpv = fmaf(fc_cp_w[k], hcf[k], fmaf(fc_cp_w[16 + k], hcb[k], cpv));
  }
  float o = fc_fin_b[0];
#pragma unroll
  for (int jj = 0; jj < 30; ++jj) o = fmaf(fc_fin_w[jj], statics[b * 30 + jj], o);
  o = fmaf(fc_fin_w[30], dpv, fmaf(fc_fin_w[31], cpv, o));
  out[b] = o;
}

extern "C" void kernel_launch(void* const* d_in, const int* in_sizes, int n_in,
                              void* d_out, int out_size, void* d_ws, size_t ws_size,
                              hipStream_t stream) {
  // setup_inputs() flattened order:
  // 0 static | 1 dp | 2 cp | 3 dp_times | 4 cp_times | 5 embed_dp | 6 embed_cp |
  // 7..11 gru_dp_fw{W_ih,W_hh,b_ih,b_hh,decays} | 12..16 gru_dp_bw |
  // 17..21 gru_cp_fw | 22..26 gru_cp_bw |
  // 27 fc_dp_w 28 fc_dp_b 29 fc_cp_w 30 fc_cp_b 31 fc_final_w 32 fc_final_b
  GruArgs ga;
  ga.embed[0] = (const float*)d_in[5];
  ga.embed[1] = (const float*)d_in[6];
  ga.idx[0]   = (const int*)d_in[1];
  ga.idx[1]   = (const int*)d_in[2];
  ga.times[0] = (const float*)d_in[3];
  ga.times[1] = (const float*)d_in[4];
  for (int g = 0; g < 4; ++g) {
    const int base = 7 + g * 5;
    ga.W_ih[g] = (const float*)d_in[base + 0];
    ga.W_hh[g] = (const float*)d_in[base + 1];
    ga.b_ih[g] = (const float*)d_in[base + 2];
    ga.b_hh[g] = (const float*)d_in[base + 3];
    ga.dec[g]  = (const float*)d_in[base + 4];
  }
  ga.hT = (float*)d_ws;  // 4*1024*16 floats = 256 KB

  dim3 grid(NB / 16, 4);
  gru_scan_kernel<<<grid, 32, 0, stream>>>(ga);

  head_kernel<<<(NB + 255) / 256, 256, 0, stream>>>(
      (const float*)d_ws, (const float*)d_in[0],
      (const float*)d_in[27], (const float*)d_in[28],
      (const float*)d_in[29], (const float*)d_in[30],
      (const float*)d_in[31], (const float*)d_in[32],
      (float*)d_out);
}